// RGCNModel_88184268522160
// MI455X (gfx1250) — compile-verified
//
#include <hip/hip_runtime.h>
#include <math.h>

typedef __attribute__((ext_vector_type(2))) float v2f;
typedef __attribute__((ext_vector_type(8))) float v8f;

#define NNODES 50000
#define NEDGES 800000
#define NREL   8
#define DF     128     // DIN == DH == 128
#define DOUTF  64
#define BN_EPS 1e-5f

// ---------------------------------------------------------------- utilities

__global__ __launch_bounds__(256) void k_zero4(float* __restrict__ p, int n4) {
  int i = blockIdx.x * blockDim.x + threadIdx.x;
  if (i < n4) ((float4*)p)[i] = make_float4(0.f, 0.f, 0.f, 0.f);
}

// per-(relation,node) incoming-edge counts (float; exact for counts < 2^24)
__global__ __launch_bounds__(256) void k_count(const int* __restrict__ ei,
                                               const int* __restrict__ et,
                                               float* __restrict__ cnt) {
  int e = blockIdx.x * blockDim.x + threadIdx.x;
  if (e >= NEDGES) return;
  int dst = ei[NEDGES + e];
  int r   = et[e];
  atomicAdd(&cnt[(size_t)r * NNODES + dst], 1.0f);
}

__global__ __launch_bounds__(256) void k_invert(float* __restrict__ inv, int n) {
  int i = blockIdx.x * blockDim.x + threadIdx.x;
  if (i < n) inv[i] = 1.0f / fmaxf(inv[i], 1.0f);
}

// one wave per edge; lanes cover the 128-float feature row (float4 each)
__global__ __launch_bounds__(256) void k_scatter(const float* __restrict__ h,
                                                 const int* __restrict__ ei,
                                                 const int* __restrict__ et,
                                                 float* __restrict__ mr, int rel) {
  int t    = blockIdx.x * blockDim.x + threadIdx.x;
  int e    = t >> 5;           // wave-uniform
  int lane = t & 31;
  if (e >= NEDGES) return;
  if (et[e] != rel) return;    // wave-uniform branch
  int src = ei[e];
  int dst = ei[NEDGES + e];
  float4 v = ((const float4*)(h + (size_t)src * DF))[lane];
  float* d = mr + (size_t)dst * DF + lane * 4;
  atomicAdd(d + 0, v.x);
  atomicAdd(d + 1, v.y);
  atomicAdd(d + 2, v.z);
  atomicAdd(d + 3, v.w);
}

// ------------------------------------------------------------- WMMA GEMM
// C[tile 16 x NCOLS] (+)= (A[16 x 128] * rowscale) @ B[128 x NCOLS]  (+ bias)
// One wave per 16-row strip; A fragment reused across all NCOLS/16 col tiles.
template <int NCOLS>
__global__ __launch_bounds__(256) void k_gemm(const float* __restrict__ A,
                                              const float* __restrict__ rowscale,
                                              const float* __restrict__ B,
                                              const float* __restrict__ bias,
                                              float* __restrict__ C,
                                              int accumulate) {
  constexpr int NT = NCOLS / 16;
  int wave = threadIdx.x >> 5;
  int lane = threadIdx.x & 31;
  int tile = blockIdx.x * 8 + wave;           // wave-uniform
  if (tile * 16 >= NNODES) return;
  int m0   = tile * 16;
  int half = lane >> 4;                        // 0: K even pair, 1: K odd pair
  int l16  = lane & 15;

  int rowA = m0 + l16;
  float scale = rowscale ? rowscale[rowA] : 1.0f;
  const float* arow = A + (size_t)rowA * DF;

  v8f acc[NT] = {};

  for (int k0 = 0; k0 < DF; k0 += 4) {
    // A 16x4 f32 fragment: v0 = {K=k0 (lanes0-15), K=k0+2 (lanes16-31)}, v1 = K+1
    float2 av = *(const float2*)(arow + k0 + 2 * half);
    v2f a;
    a.x = av.x * scale;
    a.y = av.y * scale;
#pragma unroll
    for (int j = 0; j < NT; ++j) {
      // B 4x16 f32 fragment: row K striped across lanes within a VGPR
      const float* bp = B + (size_t)(k0 + 2 * half) * NCOLS + j * 16 + l16;
      v2f b;
      b.x = bp[0];
      b.y = bp[NCOLS];
      acc[j] = __builtin_amdgcn_wmma_f32_16x16x4_f32(
          false, a, false, b, (short)0, acc[j], false, false);
    }
  }

  // C/D layout: VGPR i -> rows m0 + half*8 + i, col = j*16 + l16
#pragma unroll
  for (int j = 0; j < NT; ++j) {
    int col = j * 16 + l16;
    float bv = bias ? bias[col] : 0.0f;
#pragma unroll
    for (int i = 0; i < 8; ++i) {
      float* p = C + (size_t)(m0 + half * 8 + i) * NCOLS + col;
      float v = acc[j][i];
      if (accumulate) *p += v;
      else            *p = v + bv;
    }
  }
}

// ------------------------------------------------------------- epilogues

__global__ __launch_bounds__(256) void k_bnrelu(const float* __restrict__ a,
                                                const float* __restrict__ g,
                                                const float* __restrict__ be,
                                                const float* __restrict__ rm,
                                                const float* __restrict__ rv,
                                                float* __restrict__ o, int n) {
  int i = blockIdx.x * blockDim.x + threadIdx.x;
  if (i >= n) return;
  int c = i & (DF - 1);
  float v = (a[i] - rm[c]) * rsqrtf(rv[c] + BN_EPS) * g[c] + be[c];
  o[i] = fmaxf(v, 0.0f);
}

__global__ __launch_bounds__(256) void k_sigmoid(const float* __restrict__ a,
                                                 float* __restrict__ o, int n) {
  int i = blockIdx.x * blockDim.x + threadIdx.x;
  if (i >= n) return;
  o[i] = 1.0f / (1.0f + __expf(-a[i]));
}

// ------------------------------------------------------------- host driver

static inline int nblk(long n) { return (int)((n + 255) / 256); }

static void run_layer(const float* hin, const float* W, const float* Wroot,
                      const float* b, float* mr, const float* inv,
                      const int* ei, const int* et, float* agg, int ncols,
                      hipStream_t stream) {
  const int gemm_grid = (NNODES / 16 + 7) / 8;  // 391 blocks of 8 waves
  // root transform + bias initializes agg
  if (ncols == DF)
    k_gemm<DF><<<gemm_grid, 256, 0, stream>>>(hin, nullptr, Wroot, b, agg, 0);
  else
    k_gemm<DOUTF><<<gemm_grid, 256, 0, stream>>>(hin, nullptr, Wroot, b, agg, 0);

  for (int r = 0; r < NREL; ++r) {
    k_zero4<<<nblk((long)NNODES * DF / 4), 256, 0, stream>>>(mr, NNODES * DF / 4);
    k_scatter<<<nblk((long)NEDGES * 32), 256, 0, stream>>>(hin, ei, et, mr, r);
    const float* Wr = W + (size_t)r * DF * ncols;
    const float* iv = inv + (size_t)r * NNODES;
    if (ncols == DF)
      k_gemm<DF><<<gemm_grid, 256, 0, stream>>>(mr, iv, Wr, nullptr, agg, 1);
    else
      k_gemm<DOUTF><<<gemm_grid, 256, 0, stream>>>(mr, iv, Wr, nullptr, agg, 1);
  }
}

extern "C" void kernel_launch(void* const* d_in, const int* in_sizes, int n_in,
                              void* d_out, int out_size, void* d_ws, size_t ws_size,
                              hipStream_t stream) {
  const float* x   = (const float*)d_in[0];
  const int*   ei  = (const int*)d_in[1];
  const int*   et  = (const int*)d_in[2];
  const float* W1  = (const float*)d_in[3];
  const float* Wr1 = (const float*)d_in[4];
  const float* b1  = (const float*)d_in[5];
  const float* g1  = (const float*)d_in[6];
  const float* be1 = (const float*)d_in[7];
  const float* rm1 = (const float*)d_in[8];
  const float* rv1 = (const float*)d_in[9];
  const float* W2  = (const float*)d_in[10];
  const float* Wr2 = (const float*)d_in[11];
  const float* b2  = (const float*)d_in[12];
  const float* g2  = (const float*)d_in[13];
  const float* be2 = (const float*)d_in[14];
  const float* rm2 = (const float*)d_in[15];
  const float* rv2 = (const float*)d_in[16];
  const float* W3  = (const float*)d_in[17];
  const float* Wr3 = (const float*)d_in[18];
  const float* b3  = (const float*)d_in[19];

  // workspace carve-up (~105 MB total)
  float* inv = (float*)d_ws;                       // R*N      (1.6 MB)
  float* mr  = inv + (size_t)NREL * NNODES;        // N*128    (25.6 MB)
  float* agg = mr  + (size_t)NNODES * DF;          // N*128    (25.6 MB)
  float* h1  = agg + (size_t)NNODES * DF;          // N*128    (25.6 MB)
  float* h2  = h1  + (size_t)NNODES * DF;          // N*128    (25.6 MB)

  // per-(rel,node) 1/max(count,1)  (graph-only; shared by all 3 layers)
  k_zero4<<<nblk((long)NREL * NNODES / 4), 256, 0, stream>>>(inv, NREL * NNODES / 4);
  k_count<<<nblk(NEDGES), 256, 0, stream>>>(ei, et, inv);
  k_invert<<<nblk((long)NREL * NNODES), 256, 0, stream>>>(inv, NREL * NNODES);

  // layer 1: RGCN(128->128) + BN + ReLU
  run_layer(x, W1, Wr1, b1, mr, inv, ei, et, agg, DF, stream);
  k_bnrelu<<<nblk((long)NNODES * DF), 256, 0, stream>>>(agg, g1, be1, rm1, rv1, h1,
                                                        NNODES * DF);
  // layer 2: RGCN(128->128) + BN + ReLU
  run_layer(h1, W2, Wr2, b2, mr, inv, ei, et, agg, DF, stream);
  k_bnrelu<<<nblk((long)NNODES * DF), 256, 0, stream>>>(agg, g2, be2, rm2, rv2, h2,
                                                        NNODES * DF);
  // layer 3: RGCN(128->64) + sigmoid
  run_layer(h2, W3, Wr3, b3, mr, inv, ei, et, agg, DOUTF, stream);
  k_sigmoid<<<nblk((long)NNODES * DOUTF), 256, 0, stream>>>(agg, (float*)d_out,
                                                            NNODES * DOUTF);
}